// FineRefinement_83597243449978
// MI455X (gfx1250) — compile-verified
//
#include <hip/hip_runtime.h>
#include <hip/hip_bf16.h>
#include <math.h>

typedef __attribute__((ext_vector_type(2))) float v2f;
typedef __attribute__((ext_vector_type(8))) float v8f;

#define CCH 128          // channels
#define HF 256
#define WF 256
#define HWSZ ((size_t)HF * WF)

// ---------------------------------------------------------------------------
// Kernel 1: G = W^T * W   (128x128x128 f32 GEMM via V_WMMA_F32_16X16X4_F32)
// grid (8,8), block 32 (one wave per 16x16 tile of G)
// A[m,kk] = W[k+kk, c0+m], B[kk,n] = W[k+kk, c1+n]
// ---------------------------------------------------------------------------
__global__ void __launch_bounds__(32)
gram_wmma_kernel(const float* __restrict__ W, float* __restrict__ G) {
    const int lane = threadIdx.x & 31;
    const int m = lane & 15;        // M (and N) index within tile
    const int kh = lane >> 4;       // K half: lanes 16-31 hold K=2,3
    const int c0 = blockIdx.x * 16; // rows of G
    const int c1 = blockIdx.y * 16; // cols of G

    v8f acc = {};
    for (int k = 0; k < CCH; k += 4) {
        const int ka = k + 2 * kh;
        v2f a, b;
        a.x = W[(size_t)(ka + 0) * CCH + c0 + m];
        a.y = W[(size_t)(ka + 1) * CCH + c0 + m];
        b.x = W[(size_t)(ka + 0) * CCH + c1 + m];
        b.y = W[(size_t)(ka + 1) * CCH + c1 + m];
        acc = __builtin_amdgcn_wmma_f32_16x16x4_f32(
            false, a, false, b, (short)0, acc, false, false);
    }
    // C/D layout: VGPR r -> row M = r + 8*kh, col N = lane&15
    #pragma unroll
    for (int r = 0; r < 8; ++r) {
        const int row = r + 8 * kh;
        G[(size_t)(c0 + row) * CCH + c1 + m] = acc[r];
    }
}

// ---------------------------------------------------------------------------
// Kernel 2: Z[n, :] = q_raw[n, :] * G     (N x 128 x 128 GEMM, query gathered
// straight from fine_feat0 at (b_ids[n], :, y0, x0))
// block = 128 threads (4 waves), each wave does one 16-row tile; grid.y = 8
// ---------------------------------------------------------------------------
__global__ void __launch_bounds__(128)
query_proj_wmma_kernel(const float* __restrict__ fine0,
                       const float* __restrict__ G,
                       const float* __restrict__ mkpts0,
                       const int* __restrict__ b_ids,
                       float* __restrict__ Z, int N) {
    const int wave = threadIdx.x >> 5;
    const int lane = threadIdx.x & 31;
    const int m = lane & 15;
    const int kh = lane >> 4;
    const int n0 = (blockIdx.x * 4 + wave) * 16;
    const int c1 = blockIdx.y * 16;

    int n = n0 + m;
    if (n > N - 1) n = N - 1;  // clamp reads only; EXEC stays all-ones for WMMA

    const float mx = mkpts0[2 * n + 0];
    const float my = mkpts0[2 * n + 1];
    int x0 = (int)floorf(mx * 4.0f);
    int y0 = (int)floorf(my * 4.0f);
    x0 = min(max(x0, 0), WF - 1);
    y0 = min(max(y0, 0), HF - 1);
    const int b = b_ids[n];
    const float* qbase = fine0 + (size_t)b * CCH * HWSZ + (size_t)y0 * WF + x0;

    v8f acc = {};
    for (int k = 0; k < CCH; k += 4) {
        const int ka = k + 2 * kh;
        v2f a, bb;
        a.x  = qbase[(size_t)(ka + 0) * HWSZ];          // A[m,kk] = q_raw[n, k+kk]
        a.y  = qbase[(size_t)(ka + 1) * HWSZ];
        bb.x = G[(size_t)(ka + 0) * CCH + c1 + m];      // B[kk,n] = G[k+kk, c1+n]
        bb.y = G[(size_t)(ka + 1) * CCH + c1 + m];
        acc = __builtin_amdgcn_wmma_f32_16x16x4_f32(
            false, a, false, bb, (short)0, acc, false, false);
    }
    #pragma unroll
    for (int r = 0; r < 8; ++r) {
        const int row = r + 8 * kh;
        if (n0 + row < N)
            Z[(size_t)(n0 + row) * CCH + c1 + m] = acc[r];
    }
}

// ---------------------------------------------------------------------------
// Kernel 3: per-point 5x5 correlation (z . window), masked softmax, spatial
// expectation, final outputs. One wave32 per point, 8 waves per block.
// lanes 0..24 -> window taps; lanes 25..31 inert (-inf in the softmax).
// ---------------------------------------------------------------------------
__global__ void __launch_bounds__(256)
corr_softmax_kernel(const float* __restrict__ fine1,
                    const float* __restrict__ mkpts0,
                    const float* __restrict__ mkpts1,
                    const int* __restrict__ b_ids,
                    const float* __restrict__ Z,
                    float* __restrict__ out, int N) {
    const int wave = threadIdx.x >> 5;
    const int lane = threadIdx.x & 31;
    const int n = blockIdx.x * 8 + wave;
    if (n >= N) return;

    const float m1x = mkpts1[2 * n + 0];
    const float m1y = mkpts1[2 * n + 1];
    const int x1 = (int)floorf(m1x * 4.0f);   // NOT clipped (matches reference)
    const int y1 = (int)floorf(m1y * 4.0f);
    const int b = b_ids[n];

    float corr = -INFINITY;
    int xx = 0, yy = 0;
    if (lane < 25) {
        const int dy = lane / 5 - 2;
        const int dx = lane % 5 - 2;
        yy = y1 + dy;
        xx = x1 + dx;
        const bool valid = (yy >= 0) & (yy < HF) & (xx >= 0) & (xx < WF);
        const int yc = min(max(yy, 0), HF - 1);
        const int xc = min(max(xx, 0), WF - 1);
        const float* wbase = fine1 + (size_t)b * CCH * HWSZ + (size_t)yc * WF + xc;
        const float* z = Z + (size_t)n * CCH;
        float acc = 0.0f;
        #pragma unroll 8
        for (int c = 0; c < CCH; ++c)
            acc = fmaf(z[c], wbase[(size_t)c * HWSZ], acc);
        corr = valid ? acc : -INFINITY;
    }

    // wave32 max
    float mx = corr;
    #pragma unroll
    for (int o = 16; o > 0; o >>= 1)
        mx = fmaxf(mx, __shfl_xor(mx, o, 32));

    // exp; -inf taps (and lanes >= 25) contribute exactly 0
    const float e = (corr == -INFINITY) ? 0.0f : __expf(corr - mx);
    float s = e;
    float sx = e * (float)xx;
    float sy = e * (float)yy;
    #pragma unroll
    for (int o = 16; o > 0; o >>= 1) {
        s  += __shfl_xor(s,  o, 32);
        sx += __shfl_xor(sx, o, 32);
        sy += __shfl_xor(sy, o, 32);
    }

    if (lane == 0) {
        const float inv = 1.0f / s;
        const float offx = sx * inv - (float)x1;
        const float offy = sy * inv - (float)y1;
        const float m0x = mkpts0[2 * n + 0];
        const float m0y = mkpts0[2 * n + 1];
        // out = [mkpts0_f (N,2) | mkpts1_f (N,2) | offsets (N,2)]
        out[2 * n + 0]         = m0x * 8.0f;
        out[2 * n + 1]         = m0y * 8.0f;
        out[2 * N + 2 * n + 0] = m1x * 8.0f + offx * 2.0f;
        out[2 * N + 2 * n + 1] = m1y * 8.0f + offy * 2.0f;
        out[4 * N + 2 * n + 0] = offx;
        out[4 * N + 2 * n + 1] = offy;
    }
}

extern "C" void kernel_launch(void* const* d_in, const int* in_sizes, int n_in,
                              void* d_out, int out_size, void* d_ws, size_t ws_size,
                              hipStream_t stream) {
    const float* fine0 = (const float*)d_in[0];
    const float* fine1 = (const float*)d_in[1];
    const float* m0    = (const float*)d_in[2];
    const float* m1    = (const float*)d_in[3];
    const int*   bids  = (const int*)d_in[4];
    const float* Wp    = (const float*)d_in[5];
    const int N = in_sizes[2] / 2;

    float* G = (float*)d_ws;                                   // 128*128*4 = 64 KB
    float* Z = (float*)((char*)d_ws + (size_t)CCH * CCH * 4);  // N*128*4   = 4 MB

    // 1) Gram matrix G = W^T W
    dim3 gA(CCH / 16, CCH / 16);
    gram_wmma_kernel<<<gA, 32, 0, stream>>>(Wp, G);

    // 2) Z = gathered-query * G
    const int ntiles = (N + 15) / 16;
    const int nbx = (ntiles + 3) / 4;
    dim3 gB(nbx, CCH / 16);
    query_proj_wmma_kernel<<<gB, 128, 0, stream>>>(fine0, G, m0, bids, Z, N);

    // 3) correlation + softmax + expectation
    const int nbc = (N + 7) / 8;
    corr_softmax_kernel<<<nbc, 256, 0, stream>>>(fine1, m0, m1, bids, Z,
                                                 (float*)d_out, N);
}